// InteractionEmbedding_15375982920237
// MI455X (gfx1250) — compile-verified
//
#include <hip/hip_runtime.h>
#include <hip/hip_bf16.h>
#include <stdint.h>

// InteractionEmbedding: out[p,:] = normalize(W1T[idx1[p],:] * W2T[idx2[p],:])
// Pure bandwidth problem: 512MB of stores dominates (~22us floor @ 23.3TB/s).
// Strategy: transpose W into ws (rows contiguous, L2-resident), 1 wave32/pair,
// async-to-LDS double-buffered gather (ASYNCcnt path), NT stores for output.

typedef float v4f __attribute__((ext_vector_type(4)));
typedef int   v4i __attribute__((ext_vector_type(4)));
typedef __attribute__((address_space(1))) v4i* as1_v4i_ptr;  // global
typedef __attribute__((address_space(3))) v4i* as3_v4i_ptr;  // LDS

#define WAVES 8      // 256 threads/block
#define TILE 4       // pairs per wave per pipeline stage
#define D 128        // embedding dim (reference fixes d=128)

#if defined(__has_builtin)
#if __has_builtin(__builtin_amdgcn_global_load_async_to_lds_b128) && \
    __has_builtin(__builtin_amdgcn_s_wait_asynccnt)
#define USE_ASYNC 1
#endif
#endif
#ifndef USE_ASYNC
#define USE_ASYNC 0
#endif

__device__ __forceinline__ float wave_reduce_add(float v) {
  v += __shfl_xor(v, 16, 32);
  v += __shfl_xor(v, 8, 32);
  v += __shfl_xor(v, 4, 32);
  v += __shfl_xor(v, 2, 32);
  v += __shfl_xor(v, 1, 32);
  return v;
}

// Load a tile's 4 contiguous indices; vector b128 load when fully in-bounds.
__device__ __forceinline__ v4i load_idx4(const int* __restrict__ idx, int base, int P) {
  if (base + TILE <= P) {
    return *(const v4i*)(idx + base);     // one global_load_b128
  }
  v4i r;
#pragma unroll
  for (int j = 0; j < TILE; ++j) {
    int p = base + j;
    r[j] = (p < P) ? idx[p] : 0;          // clamp; masked at store time
  }
  return r;
}

// ---- one-time transpose: in[rows x cols] (row major) -> out[cols x rows] ----
__global__ void ie_transpose_kernel(const float* __restrict__ in,
                                    float* __restrict__ out,
                                    int rows, int cols) {
  __shared__ float tile[32][33];
  int x = blockIdx.x * 32 + threadIdx.x;  // col index in `in`
  int y0 = blockIdx.y * 32;               // row base in `in`
  for (int j = threadIdx.y; j < 32; j += 8) {
    int y = y0 + j;
    tile[j][threadIdx.x] = (y < rows && x < cols) ? in[(size_t)y * cols + x] : 0.0f;
  }
  __syncthreads();
  int ox = y0 + threadIdx.x;              // output col (= input row)
  for (int j = threadIdx.y; j < 32; j += 8) {
    int oy = blockIdx.x * 32 + j;         // output row (= input col)
    if (oy < cols && ox < rows)
      out[(size_t)oy * rows + ox] = tile[threadIdx.x][j];
  }
}

// ---- main kernel: one wave32 per pair, lane l owns float4 #l of the row ----
__global__ void __launch_bounds__(256)
ie_gather_norm_kernel(const int* __restrict__ idxA, const int* __restrict__ idxB,
                      const float* __restrict__ T1, const float* __restrict__ T2,
                      float* __restrict__ out, int P, int total_tiles) {
  const int lane  = threadIdx.x & 31;
  const int wave  = threadIdx.x >> 5;
  const int gwave = blockIdx.x * WAVES + wave;
  const int stride = gridDim.x * WAVES;

#if USE_ASYNC
  // Per-wave private double-buffered slab: 2 bufs * TILE pairs * 256 floats.
  __shared__ float smem[WAVES * 2 * TILE * 2 * D];  // 64 KB
  float* slab = smem + wave * (2 * TILE * 2 * D);

  auto issue = [&](int tileIdx, int buf) {
    float* bufp = slab + buf * (TILE * 2 * D);
    int base = tileIdx * TILE;
    v4i ia = load_idx4(idxA, base, P);
    v4i ib = load_idx4(idxB, base, P);
#pragma unroll
    for (int j = 0; j < TILE; ++j) {
      const float* g1 = T1 + (size_t)ia[j] * D + lane * 4;
      const float* g2 = T2 + (size_t)ib[j] * D + lane * 4;
      float* l1 = bufp + j * (2 * D) + lane * 4;
      float* l2 = l1 + D;
      // async DMA: 32 lanes x 16B = one full 512B row per instruction
      __builtin_amdgcn_global_load_async_to_lds_b128(
          (as1_v4i_ptr)(uintptr_t)g1,
          (as3_v4i_ptr)(uint32_t)(uintptr_t)l1,
          0, 0);
      __builtin_amdgcn_global_load_async_to_lds_b128(
          (as1_v4i_ptr)(uintptr_t)g2,
          (as3_v4i_ptr)(uint32_t)(uintptr_t)l2,
          0, 0);
    }
  };

  auto compute = [&](int tileIdx, int buf) {
    const float* bufp = slab + buf * (TILE * 2 * D);
#pragma unroll
    for (int j = 0; j < TILE; ++j) {
      int p = tileIdx * TILE + j;
      if (p >= P) break;                            // wave-uniform
      v4f a = *(const v4f*)(bufp + j * (2 * D) + lane * 4);
      v4f b = *(const v4f*)(bufp + j * (2 * D) + D + lane * 4);
      v4f m = a * b;
      float ss = m.x * m.x + m.y * m.y + m.z * m.z + m.w * m.w;
      ss = wave_reduce_add(ss);
      float sc = 1.0f / fmaxf(sqrtf(ss), 1e-12f);
      v4f o = m * sc;
      __builtin_nontemporal_store(o, (v4f*)(out + (size_t)p * D) + lane);
    }
  };

  int tile = gwave;
  int buf = 0;
  if (tile < total_tiles) issue(tile, buf);
  for (; tile < total_tiles; tile += stride) {
    int nxt = tile + stride;
    bool hasNext = nxt < total_tiles;
    if (hasNext) {
      __builtin_prefetch(idxA + (size_t)nxt * TILE, 0, 0);
      __builtin_prefetch(idxB + (size_t)nxt * TILE, 0, 0);
      issue(nxt, buf ^ 1);
      __builtin_amdgcn_s_wait_asynccnt(2 * TILE);   // older batch done
    } else {
      __builtin_amdgcn_s_wait_asynccnt(0);
    }
    asm volatile("" ::: "memory");                  // order LDS reads after wait
    compute(tile, buf);
    buf ^= 1;
  }
#else
  // Fallback: direct-to-VGPR gather (still fully coalesced 128b loads).
  for (int tile = gwave; tile < total_tiles; tile += stride) {
    int nxt = tile + stride;
    if (nxt < total_tiles) {
      __builtin_prefetch(idxA + (size_t)nxt * TILE, 0, 0);
      __builtin_prefetch(idxB + (size_t)nxt * TILE, 0, 0);
    }
    int base = tile * TILE;
    v4i ia = load_idx4(idxA, base, P);
    v4i ib = load_idx4(idxB, base, P);
#pragma unroll
    for (int j = 0; j < TILE; ++j) {
      int p = base + j;
      if (p >= P) break;                            // wave-uniform
      v4f a = *(const v4f*)(T1 + (size_t)ia[j] * D + lane * 4);
      v4f b = *(const v4f*)(T2 + (size_t)ib[j] * D + lane * 4);
      v4f m = a * b;
      float ss = m.x * m.x + m.y * m.y + m.z * m.z + m.w * m.w;
      ss = wave_reduce_add(ss);
      float sc = 1.0f / fmaxf(sqrtf(ss), 1e-12f);
      v4f o = m * sc;
      __builtin_nontemporal_store(o, (v4f*)(out + (size_t)p * D) + lane);
    }
  }
#endif
}

extern "C" void kernel_launch(void* const* d_in, const int* in_sizes, int n_in,
                              void* d_out, int out_size, void* d_ws, size_t ws_size,
                              hipStream_t stream) {
  const int*   pairs = (const int*)d_in[0];    // [2, P] int32
  const float* W1    = (const float*)d_in[3];  // [d, n1]
  const float* W2    = (const float*)d_in[4];  // [d, n2]
  float*       out   = (float*)d_out;          // [P, d]

  int P  = in_sizes[0] / 2;
  int d  = (P > 0) ? (out_size / P) : D;       // expect 128
  int n1 = in_sizes[3] / d;
  int n2 = in_sizes[4] / d;

  float* T1 = (float*)d_ws;                    // [n1, d]
  float* T2 = T1 + (size_t)n1 * d;             // [n2, d]

  dim3 tb(32, 8);
  dim3 tg1((n1 + 31) / 32, (d + 31) / 32);
  dim3 tg2((n2 + 31) / 32, (d + 31) / 32);
  ie_transpose_kernel<<<tg1, tb, 0, stream>>>(W1, T1, d, n1);
  ie_transpose_kernel<<<tg2, tb, 0, stream>>>(W2, T2, d, n2);

  int total_tiles = (P + TILE - 1) / TILE;
  int blocks = (total_tiles + WAVES - 1) / WAVES;
  if (blocks > 4096) blocks = 4096;            // ~8 tiles/wave -> pipeline depth
  if (blocks < 1) blocks = 1;
  ie_gather_norm_kernel<<<blocks, 256, 0, stream>>>(
      pairs, pairs + P, T1, T2, out, P, total_tiles);
}